// DKGE_Online_20186346291261
// MI455X (gfx1250) — compile-verified
//
#include <hip/hip_runtime.h>
#include <math.h>

typedef __attribute__((ext_vector_type(2))) float v2f;
typedef __attribute__((ext_vector_type(8))) float v8f;

#define DIMS 128
#define CTX  30
#define NP1  31   // C+1 nodes
#define PAD  32   // padded node count for WMMA tiles

// One workgroup (128 threads = 4 wave32) per (sample b, subgraph sg).
// sg: 0=ph 1=pt 2=nh 3=nt (entity), 4=pr 5=nr (relation)
__global__ __launch_bounds__(128) void dkge_subgraph_kernel(
    const int* __restrict__ pos_h, const int* __restrict__ pos_r,
    const int* __restrict__ pos_t, const int* __restrict__ neg_h,
    const int* __restrict__ neg_r, const int* __restrict__ neg_t,
    const int* __restrict__ ph_adj, const int* __restrict__ pt_adj,
    const int* __restrict__ nh_adj, const int* __restrict__ nt_adj,
    const int* __restrict__ pr_adj, const int* __restrict__ nr_adj,
    const float* __restrict__ ph_A, const float* __restrict__ pr_A,
    const float* __restrict__ pt_A, const float* __restrict__ nh_A,
    const float* __restrict__ nr_A, const float* __restrict__ nt_A,
    const float* __restrict__ entity_emb, const float* __restrict__ relation_emb,
    const float* __restrict__ entity_context, const float* __restrict__ relation_context,
    const float* __restrict__ W_ent, const float* __restrict__ W_rel,
    const float* __restrict__ gate_ent, const float* __restrict__ gate_rel,
    const float* __restrict__ v_ent, const float* __restrict__ v_rel,
    float* __restrict__ ws, int B)
{
    const int b   = blockIdx.x;
    const int sg  = blockIdx.y;          // uniform per block
    const bool isRel = (sg >= 4);

    const int*   idx = nullptr;
    const int*   adj = nullptr;
    const float* Amat = nullptr;
    switch (sg) {
        case 0: idx = pos_h; adj = ph_adj; Amat = ph_A; break;
        case 1: idx = pos_t; adj = pt_adj; Amat = pt_A; break;
        case 2: idx = neg_h; adj = nh_adj; Amat = nh_A; break;
        case 3: idx = neg_t; adj = nt_adj; Amat = nt_A; break;
        case 4: idx = pos_r; adj = pr_adj; Amat = pr_A; break;
        default: idx = neg_r; adj = nr_adj; Amat = nr_A; break;
    }
    const float* W    = isRel ? W_rel    : W_ent;
    const float* gate = isRel ? gate_rel : gate_ent;
    const float* vvec = isRel ? v_rel    : v_ent;

    __shared__ float sH [PAD][DIMS];   // 16 KB  concat(center, ctx), zero-padded row 31
    __shared__ float sA [PAD][PAD];    //  4 KB  padded adjacency
    __shared__ float sAH[PAD][DIMS];   // 16 KB  A @ H
    __shared__ float sH2[PAD][DIMS];   // 16 KB  relu(AH @ W)
    __shared__ float sO [DIMS];
    __shared__ float sScore[PAD];
    __shared__ float sRed[2];

    const int tid = threadIdx.x;       // 0..127, one feature dim per thread

    // warm L2/WGP$ for this sample's adjacency block while we gather
    __builtin_prefetch(&Amat[(size_t)b * NP1 * NP1], 0, 0);

    // ---- gather phase --------------------------------------------------
    const int center = idx[b];
    const float* centerTab = isRel ? relation_emb : entity_emb;
    const float oval = centerTab[(size_t)center * DIMS + tid];
    sO[tid]    = oval;
    sH[0][tid] = oval;
    sH[NP1][tid] = 0.0f;               // zero pad row 31

    if (!isRel) {
        // async gather: global row -> LDS directly (ASYNCcnt), no VGPR round trip
        #pragma unroll 6
        for (int c = 0; c < CTX; ++c) {
            const int a = adj[(size_t)b * CTX + c];
            const unsigned long long gaddr =
                (unsigned long long)(const void*)&entity_context[(size_t)a * DIMS + tid];
            const unsigned lds_off = (unsigned)(uintptr_t)&sH[1 + c][tid];
            asm volatile("global_load_async_to_lds_b32 %0, %1, off"
                         :: "v"(lds_off), "v"(gaddr) : "memory");
        }
    } else {
        // relation context arrives in pairs and must be summed: VALU path
        #pragma unroll 5
        for (int c = 0; c < CTX; ++c) {
            const int a0 = adj[(size_t)b * (2 * CTX) + 2 * c];
            const int a1 = adj[(size_t)b * (2 * CTX) + 2 * c + 1];
            sH[1 + c][tid] = relation_context[(size_t)a0 * DIMS + tid]
                           + relation_context[(size_t)a1 * DIMS + tid];
        }
    }

    // padded 32x32 adjacency
    for (int i = tid; i < PAD * PAD; i += 128) {
        const int r = i >> 5, c = i & 31;
        sA[r][c] = (r < NP1 && c < NP1)
                 ? Amat[(size_t)b * NP1 * NP1 + r * NP1 + c] : 0.0f;
    }

    if (!isRel)
        asm volatile("s_wait_asynccnt 0" ::: "memory");
    __syncthreads();

    // ---- WMMA fragment lane mapping (f32 16x16x4, ISA 7.12.2) ----------
    const int lane  = tid & 31;
    // wave-uniform wave id: keeps tile loops scalar, EXEC untouched around WMMA
    const int wave  = __builtin_amdgcn_readfirstlane(tid >> 5);
    const int lh    = lane & 15;           // row/col within half
    const int kOff  = (lane < 16) ? 0 : 2; // A/B: K = kOff + vgpr_idx
    const int mOff  = (lane < 16) ? 0 : 8; // C/D: M = mBase + r + mOff

    // ---- GEMM1: AH = A @ H   (32x128 = 32x32 @ 32x128) -----------------
    for (int t = wave; t < 16; t += 4) {   // 2 M-tiles x 8 N-tiles
        const int mBase = (t >> 3) << 4;
        const int nBase = (t & 7) << 4;
        v8f acc = {};
        #pragma unroll
        for (int k = 0; k < PAD; k += 4) {
            v2f a, bb;
            a.x  = sA[mBase + lh][k + kOff];
            a.y  = sA[mBase + lh][k + kOff + 1];
            bb.x = sH[k + kOff    ][nBase + lh];
            bb.y = sH[k + kOff + 1][nBase + lh];
            acc = __builtin_amdgcn_wmma_f32_16x16x4_f32(
                      false, a, false, bb, (short)0, acc, false, false);
        }
        #pragma unroll
        for (int r = 0; r < 8; ++r)
            sAH[mBase + mOff + r][nBase + lh] = acc[r];
    }
    __syncthreads();

    // ---- GEMM2: H2 = relu(AH @ W)  (32x128 = 32x128 @ 128x128) --------
    for (int t = wave; t < 16; t += 4) {
        const int mBase = (t >> 3) << 4;
        const int nBase = (t & 7) << 4;
        v8f acc = {};
        #pragma unroll
        for (int k = 0; k < DIMS; k += 4) {
            v2f a, bb;
            a.x  = sAH[mBase + lh][k + kOff];
            a.y  = sAH[mBase + lh][k + kOff + 1];
            bb.x = W[(size_t)(k + kOff    ) * DIMS + nBase + lh];
            bb.y = W[(size_t)(k + kOff + 1) * DIMS + nBase + lh];
            acc = __builtin_amdgcn_wmma_f32_16x16x4_f32(
                      false, a, false, bb, (short)0, acc, false, false);
        }
        #pragma unroll
        for (int r = 0; r < 8; ++r) {
            const float v = acc[r];
            sH2[mBase + mOff + r][nBase + lh] = v > 0.0f ? v : 0.0f;
        }
    }
    __syncthreads();

    // ---- attention: score_i = relu(H2_i * o) . v  ----------------------
    if (tid < NP1) {
        float s = 0.0f;
        for (int d = 0; d < DIMS; ++d) {
            const float hv = sH2[tid][d] * sO[d];
            s = fmaf(hv > 0.0f ? hv : 0.0f, vvec[d], s);
        }
        sScore[tid] = s;
    }
    __syncthreads();
    if (tid == 0) {
        float mx = sScore[0];
        for (int i = 1; i < NP1; ++i) mx = fmaxf(mx, sScore[i]);
        float sum = 0.0f;
        for (int i = 0; i < NP1; ++i) sum += __expf(sScore[i] - mx);
        sRed[0] = mx; sRed[1] = sum;
    }
    __syncthreads();
    if (tid < NP1)
        sScore[tid] = __expf(sScore[tid] - sRed[0]) / sRed[1];
    __syncthreads();

    // ---- sg = alpha @ H2 ; out = g*o + (1-g)*sg ------------------------
    float sgv = 0.0f;
    #pragma unroll 31
    for (int i = 0; i < NP1; ++i)
        sgv = fmaf(sScore[i], sH2[i][tid], sgv);
    const float g = 1.0f / (1.0f + __expf(-gate[tid]));
    ws[((size_t)sg * B + b) * DIMS + tid] = g * sO[tid] + (1.0f - g) * sgv;
}

// Final TransE scoring: out[0][b]=||ph+pr-pt||, out[1][b]=||nh+nr-nt||
__global__ __launch_bounds__(128) void dkge_score_kernel(
    const float* __restrict__ ws, float* __restrict__ out, int B)
{
    const int b   = blockIdx.x;
    const int tid = threadIdx.x;
    const float* ph = ws + ((size_t)0 * B + b) * DIMS;
    const float* pt = ws + ((size_t)1 * B + b) * DIMS;
    const float* nh = ws + ((size_t)2 * B + b) * DIMS;
    const float* nt = ws + ((size_t)3 * B + b) * DIMS;
    const float* pr = ws + ((size_t)4 * B + b) * DIMS;
    const float* nr = ws + ((size_t)5 * B + b) * DIMS;

    const float dp = ph[tid] + pr[tid] - pt[tid];
    const float dn = nh[tid] + nr[tid] - nt[tid];

    __shared__ float rp[128];
    __shared__ float rn[128];
    rp[tid] = dp * dp;
    rn[tid] = dn * dn;
    __syncthreads();
    #pragma unroll
    for (int s = 64; s > 0; s >>= 1) {
        if (tid < s) { rp[tid] += rp[tid + s]; rn[tid] += rn[tid + s]; }
        __syncthreads();
    }
    if (tid == 0) {
        out[b]     = sqrtf(rp[0]);
        out[B + b] = sqrtf(rn[0]);
    }
}

extern "C" void kernel_launch(void* const* d_in, const int* in_sizes, int n_in,
                              void* d_out, int out_size, void* d_ws, size_t ws_size,
                              hipStream_t stream) {
    (void)n_in; (void)out_size; (void)ws_size;
    const int B = in_sizes[0];

    const int*   pos_h  = (const int*)d_in[0];
    const int*   pos_r  = (const int*)d_in[1];
    const int*   pos_t  = (const int*)d_in[2];
    const int*   neg_h  = (const int*)d_in[3];
    const int*   neg_r  = (const int*)d_in[4];
    const int*   neg_t  = (const int*)d_in[5];
    const int*   ph_adj = (const int*)d_in[6];
    const int*   pt_adj = (const int*)d_in[7];
    const int*   nh_adj = (const int*)d_in[8];
    const int*   nt_adj = (const int*)d_in[9];
    const int*   pr_adj = (const int*)d_in[10];
    const int*   nr_adj = (const int*)d_in[11];
    const float* ph_A   = (const float*)d_in[12];
    const float* pr_A   = (const float*)d_in[13];
    const float* pt_A   = (const float*)d_in[14];
    const float* nh_A   = (const float*)d_in[15];
    const float* nr_A   = (const float*)d_in[16];
    const float* nt_A   = (const float*)d_in[17];
    const float* entity_emb       = (const float*)d_in[18];
    const float* relation_emb     = (const float*)d_in[19];
    const float* entity_context   = (const float*)d_in[20];
    const float* relation_context = (const float*)d_in[21];
    const float* W_ent    = (const float*)d_in[22];
    const float* W_rel    = (const float*)d_in[23];
    const float* gate_ent = (const float*)d_in[24];
    const float* gate_rel = (const float*)d_in[25];
    const float* v_ent    = (const float*)d_in[26];
    const float* v_rel    = (const float*)d_in[27];

    float* ws = (float*)d_ws;   // [6][B][128] fp32 subgraph outputs

    dim3 grid(B, 6, 1);
    dkge_subgraph_kernel<<<grid, 128, 0, stream>>>(
        pos_h, pos_r, pos_t, neg_h, neg_r, neg_t,
        ph_adj, pt_adj, nh_adj, nt_adj, pr_adj, nr_adj,
        ph_A, pr_A, pt_A, nh_A, nr_A, nt_A,
        entity_emb, relation_emb, entity_context, relation_context,
        W_ent, W_rel, gate_ent, gate_rel, v_ent, v_rel,
        ws, B);

    dkge_score_kernel<<<dim3(B), 128, 0, stream>>>(ws, (float*)d_out, B);
}